// CapsuleNetworksModel_20564303413615
// MI455X (gfx1250) — compile-verified
//
#include <hip/hip_runtime.h>

// Problem constants (from reference setup_inputs)
#define BTOT 512
#define N1   1152
#define N2   10
#define D2   16
#define D1   8
#define MK   (N2*D2)        // 160 columns per capsule-n GEMM
#define BT   16             // batch tile (WMMA M)
#define NBT  (BTOT/BT)      // 32 batch tiles
#define GY   6              // grid.y for routing pass
#define WPB  8              // waves per block
#define NCHUNK (GY*WPB)     // 48 n-chunks per batch tile
#define NPW  (N1/NCHUNK)    // 24 capsules per wave

typedef float v2f __attribute__((ext_vector_type(2)));
typedef float v8f __attribute__((ext_vector_type(8)));

__device__ __forceinline__ v8f wmma4(v2f a, v2f b, v8f c) {
  // D = A(16x4 f32) * B(4x16 f32) + C(16x16 f32)
  return __builtin_amdgcn_wmma_f32_16x16x4_f32(
      /*neg_a=*/false, a, /*neg_b=*/false, b,
      /*c_mod=*/(short)0, c, /*reuse_a=*/false, /*reuse_b=*/false);
}

// One routing pass: recompute pred via WMMA, (iter>0) agreement+logit update+
// softmax, accumulate s partials, write partials to per-chunk ws slot.
__global__ __launch_bounds__(256) void routing_pass(
    const float* __restrict__ u,      // [B, N1, D1]
    const float* __restrict__ W,      // [N1, N2, D2, D1]
    const float* __restrict__ vprev,  // [B, N2, D2]
    float* __restrict__ blog,         // [B, N1, N2]
    float* __restrict__ spart,        // [NBT, NCHUNK, BT, N2, D2]
    int iter)
{
  __shared__ float vsh[BT * MK];        // v tile for this batch tile
  __shared__ float nbuf[WPB][BT * N2];  // updated logits, per wave
  __shared__ float cbuf[WPB][BT * N2];  // softmax coefficients, per wave

  const int wave = threadIdx.x >> 5;
  const int lane = threadIdx.x & 31;
  const int hf   = lane >> 4;     // half: 0 -> rows b0+r, 1 -> rows b0+r+8
  const int lq   = lane & 15;     // within-half lane: WMMA column (=k)
  const int b0   = blockIdx.x * BT;
  const int nbase = (blockIdx.y * WPB + wave) * NPW;

  // Stage v[b0:b0+16, :, :] into LDS (reused by every n this block touches)
  if (iter > 0) {
    for (int i = threadIdx.x; i < BT * MK; i += 256)
      vsh[i] = vprev[(size_t)b0 * MK + i];
  }
  __syncthreads();

  const v8f zero8 = {0.f,0.f,0.f,0.f,0.f,0.f,0.f,0.f};
  v8f sac[N2];
  #pragma unroll
  for (int m = 0; m < N2; ++m) sac[m] = zero8;

  for (int nn = 0; nn < NPW; ++nn) {
    const int n = nbase + nn;

    // ---- A fragments: u[b0+lq, n, 0:8] split into two K=4 chunks ----
    // 16x4 f32 A layout: VGPR0 = K{0 | 2}, VGPR1 = K{1 | 3} (lanes 0-15 | 16-31)
    const float* up = u + ((size_t)(b0 + lq) * N1 + n) * D1;
    const float2 ua0 = *(const float2*)(up + hf * 2);       // d = hf*2, hf*2+1
    const float2 ua1 = *(const float2*)(up + 4 + hf * 2);   // d = 4+hf*2, ...
    const v2f A0 = {ua0.x, ua0.y};
    const v2f A1 = {ua1.x, ua1.y};

    // ---- pred fragments: one 16x16 WMMA tile per output capsule m ----
    // C/D layout: VGPR r <-> batch row b0+r (+8 for upper half), lane <-> k
    v8f pred[N2];
    const float* wp = W + (size_t)n * MK * D1;
    #pragma unroll
    for (int m = 0; m < N2; ++m) {
      const float* wr = wp + ((m * D2) + lq) * D1 + hf * 2;
      const float2 w0 = *(const float2*)(wr);       // B rows d=hf*2, hf*2+1
      const float2 w1 = *(const float2*)(wr + 4);   // B rows d=4+hf*2, ...
      const v2f B0 = {w0.x, w0.y};
      const v2f B1 = {w1.x, w1.y};
      v8f p = wmma4(A0, B0, zero8);
      pred[m] = wmma4(A1, B1, p);
    }

    if (iter > 0) {
      // ---- agreement = <pred[b,m,:], v[b,m,:]> via 16-lane reduction ----
      #pragma unroll
      for (int m = 0; m < N2; ++m) {
        #pragma unroll
        for (int r = 0; r < 8; ++r) {
          const int bb = r + hf * 8;
          float t = pred[m][r] * vsh[bb * MK + m * D2 + lq];
          t += __shfl_xor(t, 1, 32);
          t += __shfl_xor(t, 2, 32);
          t += __shfl_xor(t, 4, 32);
          t += __shfl_xor(t, 8, 32);
          if (lq == 0) {
            const size_t gi = ((size_t)(b0 + bb) * N1 + n) * N2 + m;
            const float oldb = (iter == 1) ? 0.f : blog[gi];
            const float nb = oldb + t;
            blog[gi] = nb;
            nbuf[wave][bb * N2 + m] = nb;
          }
        }
      }
      // ---- softmax over m: lane l (<16) owns batch row l ----
      if (lane < 16) {
        float row[N2];
        float mx = -3.0e38f;
        #pragma unroll
        for (int m = 0; m < N2; ++m) {
          const float x = nbuf[wave][lane * N2 + m];
          row[m] = x; mx = fmaxf(mx, x);
        }
        float ssum = 0.f;
        #pragma unroll
        for (int m = 0; m < N2; ++m) { row[m] = __expf(row[m] - mx); ssum += row[m]; }
        const float inv = 1.f / ssum;
        #pragma unroll
        for (int m = 0; m < N2; ++m) cbuf[wave][lane * N2 + m] = row[m] * inv;
      }
      // ---- s += c * pred ----
      #pragma unroll
      for (int m = 0; m < N2; ++m) {
        #pragma unroll
        for (int r = 0; r < 8; ++r) {
          const int bb = r + hf * 8;
          sac[m][r] += cbuf[wave][bb * N2 + m] * pred[m][r];
        }
      }
    } else {
      // iter 0: b=0 -> c = 1/10 uniform
      #pragma unroll
      for (int m = 0; m < N2; ++m)
        #pragma unroll
        for (int r = 0; r < 8; ++r)
          sac[m][r] += 0.1f * pred[m][r];
    }
  }

  // Deterministic partial write (no atomics): one slot per (btile, chunk)
  const size_t base = ((size_t)blockIdx.x * NCHUNK + blockIdx.y * WPB + wave) * (BT * MK);
  #pragma unroll
  for (int m = 0; m < N2; ++m) {
    #pragma unroll
    for (int r = 0; r < 8; ++r) {
      const int bb = r + hf * 8;
      spart[base + bb * MK + m * D2 + lq] = sac[m][r];
    }
  }
}

// Reduce chunk partials, squash, write v (and final output on last iter).
__global__ __launch_bounds__(160) void squash_kernel(
    const float* __restrict__ spart, float* __restrict__ vout,
    float* __restrict__ dout, int writeOut)
{
  const int b = blockIdx.x;
  const int t = threadIdx.x;          // t = m*16 + k
  const int btile = b >> 4;
  const int bb = b & 15;

  float sv = 0.f;
  for (int ch = 0; ch < NCHUNK; ++ch)
    sv += spart[((size_t)btile * NCHUNK + ch) * (BT * MK) + bb * MK + t];

  float sq = sv * sv;
  sq += __shfl_xor(sq, 1, 32);
  sq += __shfl_xor(sq, 2, 32);
  sq += __shfl_xor(sq, 4, 32);
  sq += __shfl_xor(sq, 8, 32);        // sum over k within the 16-lane group

  const float vv = (sq / (1.f + sq)) * sv * rsqrtf(sq + 1e-7f);
  const size_t idx = (size_t)b * MK + t;
  vout[idx] = vv;
  if (writeOut) dout[idx] = vv;
}

extern "C" void kernel_launch(void* const* d_in, const int* in_sizes, int n_in,
                              void* d_out, int out_size, void* d_ws, size_t ws_size,
                              hipStream_t stream) {
  const float* u = (const float*)d_in[0];   // [512, 1152, 8]
  const float* W = (const float*)d_in[1];   // [1, 1152, 10, 16, 8]
  (void)in_sizes; (void)n_in; (void)out_size; (void)ws_size;

  // ws layout: v | blog | spart
  float* v     = (float*)d_ws;                              // 512*160
  float* blog  = v + (size_t)BTOT * MK;                     // 512*1152*10
  float* spart = blog + (size_t)BTOT * N1 * N2;             // 32*48*16*160

  float* out = (float*)d_out;

  for (int iter = 0; iter < 3; ++iter) {
    routing_pass<<<dim3(NBT, GY), 256, 0, stream>>>(u, W, v, blog, spart, iter);
    squash_kernel<<<BTOT, 160, 0, stream>>>(spart, v, out, iter == 2 ? 1 : 0);
  }
}